// RoiPoolingTF_5970004541862
// MI455X (gfx1250) — compile-verified
//
#include <hip/hip_runtime.h>
#include <stdint.h>

// RoiPoolingTF on MI455X (gfx1250):
//   crop_and_resize(50x50x1024 fp32, 300 boxes, 14x14) + 2x2 maxpool -> (1,300,7,7,1024)
// Gather/memory-bound (feature map is 10 MB -> fully L2-resident). Uses the
// CDNA5 async global->LDS data mover (ASYNCcnt) to stage the 16 corner
// vectors per pooled cell, wave-private (no barriers), fully coalesced b128.

constexpr int FH = 50;
constexpr int FW = 50;
constexpr int FC = 1024;  // channels, contiguous innermost

__device__ __forceinline__ float bilin(float tl, float tr, float bl, float br,
                                       float fx, float fy) {
  // exact op order of the reference: x-lerp then y-lerp
  float top = tl + (tr - tl) * fx;
  float bot = bl + (br - bl) * fx;
  return top + (bot - top) * fy;
}

__global__ __launch_bounds__(256) void roi_pool_async(
    const float* __restrict__ feat, const float* __restrict__ boxes,
    float* __restrict__ out) {
  // 16 slots (4 pool sub-cells x 4 bilinear corners) x 1024 floats = 64 KB.
  // Each wave writes/reads only its own 128-channel slice of every slot.
  __shared__ __align__(16) float buf[16 * FC];

  const int tid = threadIdx.x;
  const int blk = blockIdx.x;
  const int box = blk / 49;
  const int cid = blk - box * 49;
  const int py = cid / 7;
  const int px = cid - py * 7;

  const float by1 = boxes[box * 4 + 0];
  const float bx1 = boxes[box * 4 + 1];
  const float by2 = boxes[box * 4 + 2];
  const float bx2 = boxes[box * 4 + 3];
  // grid step: (y2-y1)*(H-1)/(CROP-1)
  const float stepY = (by2 - by1) * (49.0f / 13.0f);
  const float stepX = (bx2 - bx1) * (49.0f / 13.0f);

  int ya[2], yb[2], xa[2], xb[2];
  float wy[2], wx[2];
#pragma unroll
  for (int d = 0; d < 2; ++d) {
    const float ys = by1 * 49.0f + (float)(2 * py + d) * stepY;
    const float xs = bx1 * 49.0f + (float)(2 * px + d) * stepX;
    const float yf = floorf(ys);
    const float xf = floorf(xs);
    wy[d] = ys - yf;
    wx[d] = xs - xf;
    int yi = (int)yf;
    yi = yi < 0 ? 0 : (yi > FH - 1 ? FH - 1 : yi);
    int xi = (int)xf;
    xi = xi < 0 ? 0 : (xi > FW - 1 ? FW - 1 : xi);
    ya[d] = yi;
    yb[d] = (yi + 1 > FH - 1) ? FH - 1 : (yi + 1);
    xa[d] = xi;
    xb[d] = (xi + 1 > FW - 1) ? FW - 1 : (xi + 1);
  }

  const int ch = tid * 4;  // 4 fp32 channels per lane -> b128 everywhere

  // ---- Issue all 16 async global->LDS copies (CDNA5 data mover path). ----
  // Per wave: 16 instructions, each lane moves 16 B. Completion is in-order
  // on ASYNCcnt, and all LDS/channel slices are wave-private, so the compute
  // below drains with staged s_wait_asynccnt and needs no workgroup barrier.
#pragma unroll
  for (int c = 0; c < 4; ++c) {
    const int dy = c >> 1, dx = c & 1;
    const int ry[4] = {ya[dy], ya[dy], yb[dy], yb[dy]};
    const int rx[4] = {xa[dx], xb[dx], xa[dx], xb[dx]};
#pragma unroll
    for (int k = 0; k < 4; ++k) {
      const float* g = feat + ((ry[k] * FW + rx[k]) << 10) + ch;
      // generic LDS pointer: low 32 bits are the raw LDS byte offset
      const uint32_t lds = (uint32_t)(uintptr_t)(&buf[((c << 2) + k) * FC + ch]);
      asm volatile("global_load_async_to_lds_b128 %0, %1, off"
                   :: "v"(lds), "v"((uint64_t)(uintptr_t)g)
                   : "memory");
    }
  }

  float4 acc;

#define CELL(cidx, waitlit, FIRST)                                           \
  {                                                                          \
    asm volatile("s_wait_asynccnt " waitlit ::: "memory");                   \
    const int dy = (cidx) >> 1, dx = (cidx) & 1;                             \
    const float fy = wy[dy], fx = wx[dx];                                    \
    const float4 tl = *(const float4*)&buf[(((cidx) << 2) + 0) * FC + ch];   \
    const float4 tr = *(const float4*)&buf[(((cidx) << 2) + 1) * FC + ch];   \
    const float4 bl = *(const float4*)&buf[(((cidx) << 2) + 2) * FC + ch];   \
    const float4 br = *(const float4*)&buf[(((cidx) << 2) + 3) * FC + ch];   \
    float4 v;                                                                \
    v.x = bilin(tl.x, tr.x, bl.x, br.x, fx, fy);                             \
    v.y = bilin(tl.y, tr.y, bl.y, br.y, fx, fy);                             \
    v.z = bilin(tl.z, tr.z, bl.z, br.z, fx, fy);                             \
    v.w = bilin(tl.w, tr.w, bl.w, br.w, fx, fy);                             \
    if (FIRST) {                                                             \
      acc = v;                                                               \
    } else {                                                                 \
      acc.x = fmaxf(acc.x, v.x);                                             \
      acc.y = fmaxf(acc.y, v.y);                                             \
      acc.z = fmaxf(acc.z, v.z);                                             \
      acc.w = fmaxf(acc.w, v.w);                                             \
    }                                                                        \
  }

  // Staged drain: after issuing 16 async ops, asynccnt<=12 means cell 0's
  // four copies landed (in-order completion), etc. Compute overlaps the
  // remaining in-flight copies.
  CELL(0, "0xC", true)
  CELL(1, "0x8", false)
  CELL(2, "0x4", false)
  CELL(3, "0x0", false)
#undef CELL

  // Output (1,300,7,7,1024), coalesced b128 store.
  float4* op = (float4*)(out + (((box * 7 + py) * 7 + px) << 10) + ch);
  *op = acc;
}

extern "C" void kernel_launch(void* const* d_in, const int* in_sizes, int n_in,
                              void* d_out, int out_size, void* d_ws,
                              size_t ws_size, hipStream_t stream) {
  (void)in_sizes; (void)n_in; (void)d_ws; (void)ws_size; (void)out_size;
  const float* feat = (const float*)d_in[0];    // (1,50,50,1024) fp32
  const float* boxes = (const float*)d_in[1];   // (1,300,4) fp32
  float* out = (float*)d_out;                   // (1,300,7,7,1024) fp32

  roi_pool_async<<<dim3(300 * 49), dim3(256), 0, stream>>>(feat, boxes, out);
}